// SIGRegLoss_57844619542832
// MI455X (gfx1250) — compile-verified
//
#include <hip/hip_runtime.h>

// ---------------------------------------------------------------------------
// SIGReg loss for MI455X (gfx1250, wave32).
//   proj = z @ normalize(slices)^T  via V_WMMA_F32_16X16X4_F32 (fp32 WMMA)
//   per-slice Gaussian-kernel sums on VALU (v_exp_f32)
// ---------------------------------------------------------------------------

typedef __attribute__((ext_vector_type(2))) float v2f;
typedef __attribute__((ext_vector_type(8))) float v8f;

#define N_PTS   256      // rows of z
#define D_DIM   768      // feature dim
#define M_SLC   1024     // slices
#define K_REF   17       // reference points

// ---------------- helpers ----------------

__device__ __forceinline__ float block_reduce_sum(float v, float* red) {
    const int tid = threadIdx.x;
    red[tid] = v;
    __syncthreads();
    for (int s = 128; s > 0; s >>= 1) {
        if (tid < s) red[tid] += red[tid + s];
        __syncthreads();
    }
    float r = red[0];
    __syncthreads();
    return r;
}

// Giles single-precision erfinv (|err| ~1e-6 rel)
__device__ __forceinline__ float erfinv_f(float x) {
    float w = -__logf((1.0f - x) * (1.0f + x));
    float p;
    if (w < 5.0f) {
        w = w - 2.5f;
        p = 2.81022636e-08f;
        p = fmaf(p, w, 3.43273939e-07f);
        p = fmaf(p, w, -3.5233877e-06f);
        p = fmaf(p, w, -4.39150654e-06f);
        p = fmaf(p, w, 0.00021858087f);
        p = fmaf(p, w, -0.00125372503f);
        p = fmaf(p, w, -0.00417768164f);
        p = fmaf(p, w, 0.246640727f);
        p = fmaf(p, w, 1.50140941f);
    } else {
        w = sqrtf(w) - 3.0f;
        p = -0.000200214257f;
        p = fmaf(p, w, 0.000100950558f);
        p = fmaf(p, w, 0.00134934322f);
        p = fmaf(p, w, -0.00367342844f);
        p = fmaf(p, w, 0.00573950773f);
        p = fmaf(p, w, -0.0076224613f);
        p = fmaf(p, w, 0.00943887047f);
        p = fmaf(p, w, 1.00167406f);
        p = fmaf(p, w, 2.83297682f);
    }
    return p * x;
}

// g_k = sqrt(2) * erfinv(2 q - 1), q = 0.01 + 0.06125 k
__device__ __forceinline__ float ref_point(int k) {
    float q = 0.01f + 0.06125f * (float)k;
    return erfinv_f(2.0f * q - 1.0f) * 1.41421356237309505f;
}

// ---------------- kernels ----------------

// Normalize each slice row to unit L2 norm (F.normalize, eps=1e-12)
__global__ __launch_bounds__(256) void norm_slices_kernel(
    const float* __restrict__ slices, float* __restrict__ a_norm) {
    __shared__ float red[256];
    const int row = blockIdx.x;
    const float* s = slices + (size_t)row * D_DIM;
    float ssq = 0.f;
    for (int c = threadIdx.x; c < D_DIM; c += 256) {
        float v = s[c];
        ssq += v * v;
    }
    float tot = block_reduce_sum(ssq, red);
    float scale = 1.0f / fmaxf(sqrtf(tot), 1e-12f);
    float* o = a_norm + (size_t)row * D_DIM;
    for (int c = threadIdx.x; c < D_DIM; c += 256) o[c] = s[c] * scale;
}

// Per-block partial sums of (z_a - z_b)^2
__global__ __launch_bounds__(256) void inv_loss_kernel(
    const float* __restrict__ za, const float* __restrict__ zb,
    float* __restrict__ partials, int n) {
    __shared__ float red[256];
    float acc = 0.f;
    for (int i = blockIdx.x * 256 + threadIdx.x; i < n; i += gridDim.x * 256) {
        float d = za[i] - zb[i];
        acc = fmaf(d, d, acc);
    }
    float t = block_reduce_sum(acc, red);
    if (threadIdx.x == 0) partials[blockIdx.x] = t;
}

// proj[z][m][n] = sum_d z[n][d] * a_norm[m][d]
// One wave32 per 16(n) x 16(m) tile, K loop over 768 in steps of 4.
// Tiles: 2 z * 16 ntiles * 64 mtiles = 2048 waves = 256 blocks * 8 waves.
__global__ __launch_bounds__(256) void gemm_proj_kernel(
    const float* __restrict__ z_a, const float* __restrict__ z_b,
    const float* __restrict__ a_norm, float* __restrict__ projT) {
    const int lane = threadIdx.x & 31;
    const int wave = threadIdx.x >> 5;
    const int tile = blockIdx.x * 8 + wave;     // 0..2047
    const int zi   = tile >> 10;                // 0 or 1
    const int rem  = tile & 1023;
    const int nt   = rem >> 6;                  // 0..15
    const int mt   = rem & 63;                  // 0..63

    const float* z = zi ? z_b : z_a;            // wave-uniform select

    // A fragment (16x4 f32): lane holds (row = lane&15, K = 2*(lane>>4)+{0,1})
    // B fragment (4x16 f32): lane holds (col = lane&15, K = 2*(lane>>4)+{0,1})
    const int r  = lane & 15;
    const int kb = (lane >> 4) << 1;
    const float* Ap = z      + (size_t)(nt * 16 + r) * D_DIM + kb;
    const float* Bp = a_norm + (size_t)(mt * 16 + r) * D_DIM + kb;

    v8f c = {0.f, 0.f, 0.f, 0.f, 0.f, 0.f, 0.f, 0.f};
#pragma unroll 8
    for (int k0 = 0; k0 < D_DIM; k0 += 4) {
        v2f av = *(const v2f*)(Ap + k0);
        v2f bv = *(const v2f*)(Bp + k0);
        c = __builtin_amdgcn_wmma_f32_16x16x4_f32(
                /*neg_a=*/false, av, /*neg_b=*/false, bv,
                /*c_mod=*/(short)0, c, /*reuse_a=*/false, /*reuse_b=*/false);
    }

    // D layout: (vgpr v, lane l) -> row M = v + 8*(l>>4), col N = l&15.
    // Here M-dim = n (z row), N-dim = m (slice). Store transposed: [z][m][n].
    const int m_g    = mt * 16 + (lane & 15);
    const int n_base = nt * 16 + ((lane >> 4) << 3);
    float* outp = projT + (size_t)zi * (M_SLC * N_PTS)
                        + (size_t)m_g * N_PTS + n_base;
#pragma unroll
    for (int v = 0; v < 8; ++v) outp[v] = c[v];
}

// One block per (z, slice): standardize 256 projections, then
//   t1 = (1/N^2) sum_ij exp(-0.5 (x_i-x_j)^2)
//   t2 = -(2/(N K)) sum_ik exp(-0.5 (x_i-g_k)^2)
__global__ __launch_bounds__(256) void sigreg_stats_kernel(
    const float* __restrict__ projT, float* __restrict__ sig_partials) {
    __shared__ float xs[256];
    __shared__ float red[256];
    __shared__ float gsh[K_REF];
    const int tid = threadIdx.x;

    const float x = projT[(size_t)blockIdx.x * N_PTS + tid];
    if (tid < K_REF) gsh[tid] = ref_point(tid);

    float mu = block_reduce_sum(x, red) * (1.0f / 256.0f);
    float d  = x - mu;
    float ss = block_reduce_sum(d * d, red);
    float sd = sqrtf(ss * (1.0f / 255.0f)) + 1e-6f;   // ddof=1 + EPS
    float v  = d / sd;
    xs[tid] = v;
    __syncthreads();

    float s1 = 0.f;
#pragma unroll 8
    for (int j = 0; j < 256; ++j) {
        float diff = v - xs[j];
        s1 += __expf(-0.5f * diff * diff);
    }
    float s2 = 0.f;
#pragma unroll
    for (int k = 0; k < K_REF; ++k) {
        float diff = v - gsh[k];
        s2 += __expf(-0.5f * diff * diff);
    }
    float partial = s1 * (1.0f / (256.0f * 256.0f))
                  - s2 * (2.0f / (256.0f * 17.0f));
    float tot = block_reduce_sum(partial, red);
    if (tid == 0) sig_partials[blockIdx.x] = tot;
}

// Sum partials, add constant t3, apply weights.
__global__ __launch_bounds__(256) void finalize_kernel(
    const float* __restrict__ inv_partials,
    const float* __restrict__ sig_partials,
    float* __restrict__ out) {
    __shared__ float red[256];
    const int tid = threadIdx.x;

    float inv_sum = block_reduce_sum(inv_partials[tid], red);

    float sacc = 0.f;
    for (int i = tid; i < 2 * M_SLC; i += 256) sacc += sig_partials[i];
    float sig_sum = block_reduce_sum(sacc, red);

    if (tid == 0) {
        float g[K_REF];
#pragma unroll
        for (int k = 0; k < K_REF; ++k) g[k] = ref_point(k);
        float t3 = 0.f;
        for (int k = 0; k < K_REF; ++k)
            for (int l = 0; l < K_REF; ++l) {
                float dd = g[k] - g[l];
                t3 += __expf(-0.5f * dd * dd);
            }
        t3 *= (1.0f / (17.0f * 17.0f));

        float inv = inv_sum * (1.0f / (float)(N_PTS * D_DIM));
        // sigreg = 0.5*(mean_a + mean_b) = 0.5*sig_sum/M + t3
        float sig = 0.5f * sig_sum * (1.0f / (float)M_SLC) + t3;
        out[0] = 25.0f * inv + 25.0f * sig;
    }
}

// ---------------- launch ----------------

extern "C" void kernel_launch(void* const* d_in, const int* in_sizes, int n_in,
                              void* d_out, int out_size, void* d_ws, size_t ws_size,
                              hipStream_t stream) {
    const float* z_a    = (const float*)d_in[0];
    const float* z_b    = (const float*)d_in[1];
    const float* slices = (const float*)d_in[2];

    float* ws = (float*)d_ws;
    float* inv_partials = ws;                       // 256 floats
    float* sig_partials = ws + 256;                 // 2048 floats
    float* a_norm       = ws + 4096;                // 1024*768 floats
    float* projT        = ws + 4096 + M_SLC * D_DIM; // 2*1024*256 floats

    norm_slices_kernel<<<M_SLC, 256, 0, stream>>>(slices, a_norm);
    inv_loss_kernel<<<256, 256, 0, stream>>>(z_a, z_b, inv_partials, N_PTS * D_DIM);
    gemm_proj_kernel<<<256, 256, 0, stream>>>(z_a, z_b, a_norm, projT);
    sigreg_stats_kernel<<<2 * M_SLC, 256, 0, stream>>>(projT, sig_partials);
    finalize_kernel<<<1, 256, 0, stream>>>(inv_partials, sig_partials, (float*)d_out);
}